// StixelLoss_53377853555244
// MI455X (gfx1250) — compile-verified
//
#include <hip/hip_runtime.h>
#include <hip/hip_bf16.h>
#include <math.h>

// ---------------------------------------------------------------------------
// StixelLoss for MI455X (gfx1250, wave32).
//   inputs : (B=256, C=2, H=192, W=256) f32   (100 MB, re-read by kernel B)
//   targets: (1, B, C, H, W)            f32   (100 MB, read once -> NT loads)
//   out    : scalar f32
// Memory-bound: ~200MB HBM traffic -> ~8.6us floor at 23.3 TB/s; targets are
// loaded non-temporal so the 100MB inputs stay resident in the 192MB L2 and
// the dense-penalty re-read of channel 1 hits L2. BCE uses the hardware
// transcendental (v_log_f32), keeping the trans pipe (~5us) under the HBM
// stream. WMMA (V_WMMA_F32_16X16X4_F32, ones-B) performs the exact f32
// cross-lane reduction in every wave.
// ---------------------------------------------------------------------------

#define B_DIM 256
#define C_DIM 2
#define H_DIM 192
#define W_DIM 256
#define HW    (H_DIM * W_DIM)                   // 49152
#define N_TOTAL (B_DIM * C_DIM * H_DIM * W_DIM) // 25,165,824
#define N4    (N_TOTAL / 4)                     // 6,291,456 float4s
#define HW4   (HW / 4)                          // 12288

#define ALPHA     1.0f
#define BETA      0.001f
#define GAMMA     0.1f
#define THRESH    0.5f
#define LOG_CLAMP -100.0f

typedef float v2f __attribute__((ext_vector_type(2)));
typedef float v4f __attribute__((ext_vector_type(4)));   // native clang vector
typedef float v8f __attribute__((ext_vector_type(8)));

// Exact f32 sum of all 32 lanes' `s` via the matrix pipe.
// A (16x4 f32): lane L<16 -> A[L][0..1], lane L>=16 -> A[L-16][2..3].
// With B = ones (4x16): D[m][n] = s[m] + s[m+16] for all n.
// Lane L<16 holds D[0..7][L] in its 8 result VGPRs, lane L>=16 holds D[8..15].
// Sum own 8 regs + shfl_xor(16) => full 32-lane sum in every lane.
__device__ __forceinline__ float wave_reduce_wmma(float s) {
    v2f a; a[0] = s;    a[1] = 0.0f;
    v2f b; b[0] = 1.0f; b[1] = 1.0f;
    v8f c = {};
    c = __builtin_amdgcn_wmma_f32_16x16x4_f32(
            /*neg_a=*/false, a, /*neg_b=*/false, b,
            /*c_mod=*/(short)0, c, /*reuse_a=*/false, /*reuse_b=*/false);
    float h = ((c[0] + c[1]) + (c[2] + c[3])) + ((c[4] + c[5]) + (c[6] + c[7]));
    h += __shfl_xor(h, 16, 32);   // combine the two half-wave row groups
    return h;
}

// Block-level accumulation (blockDim.x == 256, i.e. exactly 8 waves):
// WMMA wave reduce -> LDS -> straight-line 8-way add -> one atomicAdd.
__device__ __forceinline__ void block_accum_256(float s, float* out) {
    __shared__ float lds[8];
    float ws = wave_reduce_wmma(s);          // every wave, full EXEC
    const int lane = threadIdx.x & 31;
    const int wid  = threadIdx.x >> 5;
    if (lane == 0) lds[wid] = ws;
    __syncthreads();
    if (threadIdx.x == 0) {
        float t = ((lds[0] + lds[1]) + (lds[2] + lds[3])) +
                  ((lds[4] + lds[5]) + (lds[6] + lds[7]));
        atomicAdd(out, t);
    }
}

// ---------------------------------------------------------------------------
// Kernel A: streaming elementwise pass (BCE + cuts). 128b-vector grid-stride.
// Per element: 2x v_log_f32 (hw trans) + ~7 VALU -> stays under HBM time.
// ---------------------------------------------------------------------------
__device__ __forceinline__ float bce_elem(float p, float t) {
    float logp   = fmaxf(__logf(p),        LOG_CLAMP);   // v_log_f32 * ln2
    float log1mp = fmaxf(__logf(1.0f - p), LOG_CLAMP);
    // t*logp + (1-t)*log1mp
    return fmaf(t, logp - log1mp, log1mp);
}

__global__ void __launch_bounds__(256)
bce_cuts_kernel(const float* __restrict__ p, const float* __restrict__ t,
                float* __restrict__ out) {
    const float inv_n = 1.0f / (float)N_TOTAL;
    int tid    = blockIdx.x * blockDim.x + threadIdx.x;
    int stride = gridDim.x * blockDim.x;

    float acc_bce = 0.0f;  // sum of (t*logp + (1-t)*log1mp)
    float acc_cut = 0.0f;  // count of channel-0 elements > 0.5

    const v4f* __restrict__ p4 = (const v4f*)p;
    const v4f* __restrict__ t4 = (const v4f*)t;

    for (int i = tid; i < N4; i += stride) {
        v4f pv = p4[i];                                // RT: keep in L2
        v4f tv = __builtin_nontemporal_load(t4 + i);   // NT: read-once stream
        float s0 = bce_elem(pv[0], tv[0]);
        float s1 = bce_elem(pv[1], tv[1]);
        float s2 = bce_elem(pv[2], tv[2]);
        float s3 = bce_elem(pv[3], tv[3]);
        acc_bce += (s0 + s1) + (s2 + s3);
        // channel of these 4 contiguous elements (HW divisible by 4)
        if (((i / HW4) & 1) == 0) {
            float c0 = (pv[0] > THRESH) ? 1.0f : 0.0f;
            float c1 = (pv[1] > THRESH) ? 1.0f : 0.0f;
            float c2 = (pv[2] > THRESH) ? 1.0f : 0.0f;
            float c3 = (pv[3] > THRESH) ? 1.0f : 0.0f;
            acc_cut += (c0 + c1) + (c2 + c3);
        }
    }
    // loss contribution: -ALPHA*mean(bce) + BETA*count
    float s = acc_cut * BETA - acc_bce * (ALPHA * inv_n);
    block_accum_256(s, out);
}

// ---------------------------------------------------------------------------
// Kernel B: dense penalty. One thread per (b, w) column of channel 1,
// sequential scan over rows i = 0..H-2 (matches jnp.arange(H-1)).
// Consecutive threads read consecutive addresses -> coalesced 128B/wave;
// channel-1 data is L2-resident after kernel A (targets were NT).
// ---------------------------------------------------------------------------
__global__ void __launch_bounds__(256)
dense_kernel(const float* __restrict__ p, float* __restrict__ out) {
    int col = blockIdx.x * blockDim.x + threadIdx.x;   // 0 .. B*W-1
    int b = col >> 8;          // / W_DIM
    int w = col & (W_DIM - 1);
    // channel 1 of batch b, row 0, column w
    const float* __restrict__ base =
        p + ((size_t)b * C_DIM + 1) * (size_t)HW + (size_t)w;

    float acc = 0.0f;
    int prev = 0;              // reference quirk: row 0 never counts as prev
#pragma unroll 4
    for (int i = 0; i < H_DIM - 1; ++i) {
        float v = base[(size_t)i * W_DIM];
        if (v > THRESH) {
            if (prev != 0) {
                float d = (float)(i - prev);
                acc += 1.0f / (d * d * d);
            }
            prev = i;
        }
    }
    block_accum_256(acc * GAMMA, out);
}

// ---------------------------------------------------------------------------
extern "C" void kernel_launch(void* const* d_in, const int* in_sizes, int n_in,
                              void* d_out, int out_size, void* d_ws, size_t ws_size,
                              hipStream_t stream) {
    (void)in_sizes; (void)n_in; (void)d_ws; (void)ws_size;
    const float* inputs  = (const float*)d_in[0];
    const float* targets = (const float*)d_in[1];
    float* out = (float*)d_out;

    // zero the scalar accumulator (graph-capture safe)
    (void)hipMemsetAsync(out, 0, (size_t)out_size * sizeof(float), stream);

    // Kernel A: 6144 blocks x 256 threads -> 1.57M threads, 4 float4-iters each.
    bce_cuts_kernel<<<6144, 256, 0, stream>>>(inputs, targets, out);

    // Kernel B: one thread per (b,w) column -> 65536 threads (2048 waves).
    dense_kernel<<<(B_DIM * W_DIM) / 256, 256, 0, stream>>>(inputs, out);
}